// Attn_56495999812048
// MI455X (gfx1250) — compile-verified
//
#include <hip/hip_runtime.h>
#include <hip/hip_bf16.h>

// Problem constants (match reference).
#define H_SIZE 256
#define N_LEN  256
#define T_LEN  1024
#define BATCH  64

typedef __attribute__((ext_vector_type(2))) float v2f;
typedef __attribute__((ext_vector_type(8))) float v8f;

// ---------------------------------------------------------------------------
// Kernel 0: u1[h] = sum_i W[i, h] * v[i]; u2[h] = sum_i W[i, H+h] * v[i];
//           c = b . v.   W is (H, 2H) row-major.  One block of 256 threads.
// ---------------------------------------------------------------------------
__global__ __launch_bounds__(H_SIZE) void prep_kernel(
    const float* __restrict__ W, const float* __restrict__ bvec,
    const float* __restrict__ v, float* __restrict__ u1,
    float* __restrict__ u2, float* __restrict__ cc) {
  const int h = threadIdx.x;
  float a1 = 0.0f, a2 = 0.0f;
  for (int i = 0; i < H_SIZE; ++i) {
    const float vi = v[i];
    a1 += W[i * (2 * H_SIZE) + h] * vi;
    a2 += W[i * (2 * H_SIZE) + H_SIZE + h] * vi;
  }
  u1[h] = a1;
  u2[h] = a2;
  if (h == 0) {
    float s = 0.0f;
    for (int i = 0; i < H_SIZE; ++i) s += bvec[i] * v[i];
    cc[0] = s;
  }
}

// ---------------------------------------------------------------------------
// Kernel 1: WMMA GEMV.  src is (rows, 256) row-major with rows = outer*BATCH,
// row index m = outer_idx*64 + b.  Computes dot(src[m,:], u) for 16 rows per
// wave via V_WMMA_F32_16X16X4_F32 (B operand = u broadcast across all 16 N
// columns, so every D column equals the dot product).  Result is written
// TRANSPOSED: out[b * L + outer_idx], so the softmax kernel reads coalesced.
//
// ISA layouts used (CDNA5 ISA §7.12.2):
//   A 16x4 f32 : lane l holds rows M=l&15, K = (l>>4)*2 + {vgpr0,vgpr1}
//   B 4x16 f32 : VGPR r, lanes 0-15 -> K=r, lanes 16-31 -> K=r+2 (mirrors C/D)
//   D 16x16 f32: VGPR r, lanes 0-15 -> M=r, lanes 16-31 -> M=r+8
// ---------------------------------------------------------------------------
__global__ __launch_bounds__(256) void gemv_wmma_kernel(
    const float* __restrict__ src, const float* __restrict__ u,
    float* __restrict__ out, int L) {
  const int lane  = threadIdx.x & 31;
  const int wave  = blockIdx.x * (blockDim.x >> 5) + (threadIdx.x >> 5);
  const long m0   = (long)wave * 16;       // first of 16 rows for this wave
  const int mrow  = lane & 15;
  const int khalf = (lane >> 4) << 1;      // 0 for lanes 0-15, 2 for 16-31

  const float* arow = src + (m0 + mrow) * H_SIZE + khalf;
  const float* brow = u + khalf;

  v8f acc = {0.0f, 0.0f, 0.0f, 0.0f, 0.0f, 0.0f, 0.0f, 0.0f};
#pragma unroll 4
  for (int k0 = 0; k0 < H_SIZE; k0 += 4) {
    v2f a = *(const v2f*)(arow + k0);   // 8B load, 8B aligned
    v2f b = *(const v2f*)(brow + k0);   // u slice (L0/L2 resident)
    acc = __builtin_amdgcn_wmma_f32_16x16x4_f32(
        /*neg_a=*/false, a, /*neg_b=*/false, b,
        /*c_mod=*/(short)0, acc, /*reuse_a=*/false, /*reuse_b=*/false);
  }

  // D-matrix: lane 0 holds rows m0+0..7 in acc[0..7]; lane 16 rows m0+8..15.
  if (lane == 0) {
#pragma unroll
    for (int r = 0; r < 8; ++r) {
      const long m = m0 + r;                      // m = outer*64 + b
      out[(m & 63) * (long)L + (m >> 6)] = acc[r];
    }
  } else if (lane == 16) {
#pragma unroll
    for (int r = 0; r < 8; ++r) {
      const long m = m0 + 8 + r;
      out[(m & 63) * (long)L + (m >> 6)] = acc[r];
    }
  }
}

// ---------------------------------------------------------------------------
// Kernel 2: one block (256 threads) per (b, n) row of length T=1024.
//   e[t] = exp(tanh(sh[b,n] + se[b,t] + c));  out = e / sum_t e
// tanh is bounded, so skipping max-subtraction is numerically safe and
// mathematically identical to jax.nn.softmax.
// ---------------------------------------------------------------------------
__global__ __launch_bounds__(256) void softmax_kernel(
    const float* __restrict__ sh, const float* __restrict__ se,
    const float* __restrict__ cc, float* __restrict__ out) {
  __shared__ float red[256];
  const int row = blockIdx.x;       // b * N_LEN + n
  const int b   = row >> 8;         // N_LEN == 256
  const int tid = threadIdx.x;

  const float s = sh[row] + cc[0];
  const float4 sev = ((const float4*)(se + (long)b * T_LEN))[tid];

  const float e0 = __expf(tanhf(s + sev.x));
  const float e1 = __expf(tanhf(s + sev.y));
  const float e2 = __expf(tanhf(s + sev.z));
  const float e3 = __expf(tanhf(s + sev.w));

  red[tid] = (e0 + e1) + (e2 + e3);
  __syncthreads();
#pragma unroll
  for (int off = 128; off > 0; off >>= 1) {
    if (tid < off) red[tid] += red[tid + off];
    __syncthreads();
  }
  const float inv = 1.0f / red[0];

  const float4 o = make_float4(e0 * inv, e1 * inv, e2 * inv, e3 * inv);
  ((float4*)(out + (long)row * T_LEN))[tid] = o;
}

// ---------------------------------------------------------------------------
extern "C" void kernel_launch(void* const* d_in, const int* in_sizes, int n_in,
                              void* d_out, int out_size, void* d_ws, size_t ws_size,
                              hipStream_t stream) {
  const float* hidden = (const float*)d_in[0];  // (N, B, H)
  const float* enc    = (const float*)d_in[1];  // (T, B, H)
  const float* W      = (const float*)d_in[2];  // (H, 2H)
  const float* bvec   = (const float*)d_in[3];  // (H,)
  const float* v      = (const float*)d_in[4];  // (H,)
  float* out = (float*)d_out;                   // (B, N, T)

  // Workspace layout (floats): u1[256] | u2[256] | c[1] | pad | sh[B*N] | se[B*T]
  float* ws = (float*)d_ws;
  float* u1 = ws;
  float* u2 = ws + 256;
  float* cc = ws + 512;
  float* sh = ws + 1024;                 // BATCH * N_LEN = 16384
  float* se = ws + 1024 + BATCH * N_LEN; // BATCH * T_LEN = 65536

  prep_kernel<<<1, H_SIZE, 0, stream>>>(W, bvec, v, u1, u2, cc);

  // hidden: N*B = 16384 rows -> 1024 waves -> 128 blocks of 8 waves
  gemv_wmma_kernel<<<(N_LEN * BATCH) / 16 / 8, 256, 0, stream>>>(hidden, u1, sh, N_LEN);
  // enc: T*B = 65536 rows -> 4096 waves -> 512 blocks
  gemv_wmma_kernel<<<(T_LEN * BATCH) / 16 / 8, 256, 0, stream>>>(enc, u2, se, T_LEN);

  softmax_kernel<<<BATCH * N_LEN, 256, 0, stream>>>(sh, se, cc, out);
}